// IsometricLoss_44427141710218
// MI455X (gfx1250) — compile-verified
//
#include <hip/hip_runtime.h>
#include <hip/hip_bf16.h>
#include <math.h>
#include <stdint.h>

typedef __attribute__((ext_vector_type(2))) float v2f;
typedef __attribute__((ext_vector_type(8))) float v8f;

#define WAVES        8
#define ROWS_W       16             // rows per wave
#define ROWS_B       (WAVES*ROWS_W) // 128 rows per block
#define DDIM         128
#define MDIM         64
#define XSTR         132            // X LDS row stride (floats): 528B, 16B aligned, conflict-free frags
#define MSTR         132            // mus LDS row stride
#define RSTR         68             // r LDS row stride: 272B, 16B aligned, conflict-free epilogue reads

// LDS layout (floats):
//   mus_lds : MDIM*MSTR   = 8448
//   x_lds   : ROWS_B*XSTR = 16896
//   r_lds   : ROWS_B*RSTR = 8704
//   x2      : ROWS_B      = 128
//   m2      : MDIM        = 64
//   red     : WAVES       = 8
#define LDS_FLOATS (MDIM*MSTR + ROWS_B*XSTR + ROWS_B*RSTR + ROWS_B + MDIM + WAVES)

// Async global->LDS 16-byte copy (CDNA5, ASYNCcnt-tracked, no VGPR round-trip).
// dst_off = per-lane LDS byte offset (low 32 bits of generic shared pointer),
// src     = per-lane 64-bit global address.
__device__ __forceinline__ void async_copy_b128(uint32_t lds_byte_off, const void* gsrc) {
    asm volatile("global_load_async_to_lds_b128 %0, %1, off"
                 :: "v"(lds_byte_off), "v"((uint64_t)(uintptr_t)gsrc)
                 : "memory");
}
__device__ __forceinline__ uint32_t lds_off(const void* p) {
    return (uint32_t)(uintptr_t)p;   // generic shared ptr: low 32 bits = LDS offset
}

__global__ void __launch_bounds__(256)
isoloss_main_kernel(const float* __restrict__ X,
                    const float* __restrict__ R,
                    const float* __restrict__ MUS,
                    float* __restrict__ block_partials,
                    int N)
{
    extern __shared__ float smem[];
    float* mus_lds = smem;
    float* x_lds   = mus_lds + MDIM * MSTR;
    float* r_lds   = x_lds   + ROWS_B * XSTR;
    float* x2      = r_lds   + ROWS_B * RSTR;
    float* m2      = x2      + ROWS_B;
    float* red     = m2      + MDIM;

    const int tid   = threadIdx.x;
    const int wave  = tid >> 5;
    const int lane  = tid & 31;
    const int row0  = blockIdx.x * ROWS_B;

    // ===== async staging: mus (8/thread), X (16/thread), r (8/thread) =====
    // Issue counts are wave-uniform (OOB rows clamped, not branched), so the
    // counter-threshold wait below is exact.

    // mus: 64x128 -> padded LDS rows. 2048 16B chunks over 256 threads.
    #pragma unroll
    for (int it = 0; it < 8; ++it) {
        const int idx = tid + it * 256;          // 0..2047
        const int mr  = idx >> 5;                // 32 chunks per row
        const int c4  = idx & 31;
        async_copy_b128(lds_off(mus_lds + mr * MSTR + c4 * 4),
                        MUS + (size_t)mr * DDIM + c4 * 4);
    }

    // X: this wave's 16x128 tile, one full row (32 lanes x 16B) per issue.
    const int wrow0 = row0 + wave * ROWS_W;
    #pragma unroll
    for (int rr = 0; rr < ROWS_W; ++rr) {
        int grow = wrow0 + rr;
        grow = grow < N ? grow : N - 1;          // clamp: uniform issue, masked later
        async_copy_b128(lds_off(x_lds + (wave * ROWS_W + rr) * XSTR + lane * 4),
                        X + (size_t)grow * DDIM + lane * 4);
    }

    // r: 128x64 tile -> padded LDS rows. 2048 16B chunks over 256 threads.
    // These stay in flight across the whole WMMA GEMM.
    #pragma unroll
    for (int it = 0; it < 8; ++it) {
        const int idx = tid + it * 256;          // 0..2047
        const int rr  = idx >> 4;                // 16 chunks per row
        const int c4  = idx & 15;
        int grow = row0 + rr;
        grow = grow < N ? grow : N - 1;
        async_copy_b128(lds_off(r_lds + rr * RSTR + c4 * 4),
                        R + (size_t)grow * MDIM + c4 * 4);
    }

    // Wait for mus + X only (first 24 of 32 issued; async loads complete in
    // order, so asynccnt <= 8 implies everything except the 8 r-loads is done).
    asm volatile("s_wait_asynccnt 8" ::: "memory");
    __syncthreads();

    // ===== row norms ||x||^2 (threads 0..127) and ||mu||^2 (threads 128..191) =====
    if (tid < ROWS_B) {
        const float* rp = x_lds + tid * XSTR;
        float s = 0.f;
        #pragma unroll 8
        for (int k = 0; k < DDIM; ++k) s += rp[k] * rp[k];
        x2[tid] = s;
    } else if (tid < ROWS_B + MDIM) {
        const int mm = tid - ROWS_B;
        const float* rp = mus_lds + mm * MSTR;
        float s = 0.f;
        #pragma unroll 8
        for (int k = 0; k < DDIM; ++k) s += rp[k] * rp[k];
        m2[mm] = s;
    }
    __syncthreads();

    // ===== WMMA GEMM: 16x64 output per wave, K = 128 in steps of 4 =====
    // A (16x4 f32) frag: lane L -> M = L%16, K = 2*(L/16) + {0,1}
    // B (4x16 f32) frag: lane L -> N = L%16, K = 2*(L/16) + {0,1}  (B[k][n] = mus[n][k])
    const int mloc  = lane & 15;
    const int khalf = lane >> 4;
    const float* aptr = x_lds   + (wave * ROWS_W + mloc) * XSTR + 2 * khalf;
    const float* bptr = mus_lds + mloc * MSTR + 2 * khalf;

    v8f acc[4] = {v8f{}, v8f{}, v8f{}, v8f{}};
    #pragma unroll 4
    for (int k0 = 0; k0 < DDIM; k0 += 4) {
        const v2f a  = *reinterpret_cast<const v2f*>(aptr + k0);
        const v2f b0 = *reinterpret_cast<const v2f*>(bptr + 0 * 16 * MSTR + k0);
        const v2f b1 = *reinterpret_cast<const v2f*>(bptr + 1 * 16 * MSTR + k0);
        const v2f b2 = *reinterpret_cast<const v2f*>(bptr + 2 * 16 * MSTR + k0);
        const v2f b3 = *reinterpret_cast<const v2f*>(bptr + 3 * 16 * MSTR + k0);
        acc[0] = __builtin_amdgcn_wmma_f32_16x16x4_f32(false, a, false, b0, (short)0, acc[0], false, false);
        acc[1] = __builtin_amdgcn_wmma_f32_16x16x4_f32(false, a, false, b1, (short)0, acc[1], false, false);
        acc[2] = __builtin_amdgcn_wmma_f32_16x16x4_f32(false, a, false, b2, (short)0, acc[2], false, false);
        acc[3] = __builtin_amdgcn_wmma_f32_16x16x4_f32(false, a, false, b3, (short)0, acc[3], false, false);
    }

    // r tile must now be resident (overlapped with the GEMM above).
    asm volatile("s_wait_asynccnt 0" ::: "memory");
    __syncthreads();

    // ===== epilogue: sim -> -log(sigmoid(sim)+eps), weight by r, accumulate =====
    // C layout: element i of v8f -> M = i + 8*khalf, N = lane%16
    float pacc = 0.f;
    #pragma unroll
    for (int t = 0; t < 4; ++t) {
        const int col = t * 16 + mloc;
        const float m2v = m2[col];
        #pragma unroll
        for (int i = 0; i < 8; ++i) {
            const int rloc = wave * ROWS_W + i + 8 * khalf;
            const int grow = row0 + rloc;
            if (grow < N) {
                const float dot = acc[t][i];
                float sq = x2[rloc] + m2v - 2.f * dot;
                sq = fmaxf(sq, 0.f);
                const float d = sqrtf(sq);                 // d = -sim >= 0
                const float sg = 1.f / (1.f + expf(d));    // sigmoid(sim)
                const float simz = -logf(sg + 1e-8f);
                pacc += r_lds[rloc * RSTR + col] * simz;
            }
        }
    }

    // ===== deterministic block reduction: wave shuffles -> LDS -> thread 0 =====
    #pragma unroll
    for (int off = 16; off > 0; off >>= 1)
        pacc += __shfl_down(pacc, off, 32);
    if (lane == 0) red[wave] = pacc;
    __syncthreads();
    if (tid == 0) {
        float s = 0.f;
        #pragma unroll
        for (int w = 0; w < WAVES; ++w) s += red[w];
        block_partials[blockIdx.x] = s;
    }
}

__global__ void __launch_bounds__(256)
isoloss_reduce_kernel(const float* __restrict__ partials, int nblocks,
                      float* __restrict__ out, float invN)
{
    __shared__ float red[8];
    const int tid  = threadIdx.x;
    const int wave = tid >> 5;
    const int lane = tid & 31;

    float s = 0.f;
    for (int i = tid; i < nblocks; i += blockDim.x) s += partials[i];
    #pragma unroll
    for (int off = 16; off > 0; off >>= 1)
        s += __shfl_down(s, off, 32);
    if (lane == 0) red[wave] = s;
    __syncthreads();
    if (tid == 0) {
        float tot = 0.f;
        #pragma unroll
        for (int w = 0; w < 8; ++w) tot += red[w];
        out[0] = tot * invN;
    }
}

extern "C" void kernel_launch(void* const* d_in, const int* in_sizes, int n_in,
                              void* d_out, int out_size, void* d_ws, size_t ws_size,
                              hipStream_t stream) {
    const float* X   = (const float*)d_in[0];   // [N,128]
    const float* R   = (const float*)d_in[1];   // [N,64]
    const float* MUS = (const float*)d_in[2];   // [64,128]

    const int N = in_sizes[0] / DDIM;           // 200000
    const int nblocks = (N + ROWS_B - 1) / ROWS_B;

    float* partials = (float*)d_ws;             // nblocks floats of scratch

    const size_t shmem = (size_t)LDS_FLOATS * sizeof(float);
    isoloss_main_kernel<<<nblocks, 256, shmem, stream>>>(X, R, MUS, partials, N);
    isoloss_reduce_kernel<<<1, 256, 0, stream>>>(partials, nblocks, (float*)d_out,
                                                 1.0f / (float)N);
}